// ChunkLayer_31507880083555
// MI455X (gfx1250) — compile-verified
//
#include <hip/hip_runtime.h>
#include <hip/hip_bf16.h>
#include <stdint.h>

// Problem dims fixed by reference setup_inputs(): B=4, L=8192, D=1024
#define BB     4
#define LL     8192
#define DD     1024
#define MAXC   2048            // LL / 4
#define ROWS   4               // rows gathered per block in kernel 2

// ---------------------------------------------------------------------------
// Kernel 1: stable-partition index computation (one block per batch).
//   True tokens keep order first, false tokens after (matches stable argsort
//   of  arange(L) + (~mask)*L ).  Writes take_idx (int) and num_chunks.
// ---------------------------------------------------------------------------
__global__ __launch_bounds__(1024)
void chunk_scan_kernel(const int* __restrict__ mask,     // [BB*LL], 0/1
                       int* __restrict__ take_ws,        // [BB*MAXC]
                       int* __restrict__ nc_ws)          // [BB]
{
    __shared__ int s[1024];
    const int b = blockIdx.x;
    const int t = threadIdx.x;
    const int base = t * 8;                 // 8 elements per thread (8192/1024)

    int m[8];
    int cnt = 0;
    #pragma unroll
    for (int k = 0; k < 8; ++k) {
        m[k] = (mask[(size_t)b * LL + base + k] != 0) ? 1 : 0;
        cnt += m[k];
    }

    // Hillis-Steele inclusive scan over per-thread counts
    s[t] = cnt;
    __syncthreads();
    #pragma unroll
    for (int off = 1; off < 1024; off <<= 1) {
        int v = (t >= off) ? s[t - off] : 0;
        __syncthreads();
        s[t] += v;
        __syncthreads();
    }
    const int incl = s[t];
    const int T    = s[1023];               // total number of true tokens
    const int P    = incl - cnt;            // exclusive prefix for this thread

    int pT = P;
    #pragma unroll
    for (int k = 0; k < 8; ++k) {
        const int i   = base + k;
        const int pos = m[k] ? pT : (T + i - pT);   // stable partition position
        if (pos < MAXC)
            take_ws[(size_t)b * MAXC + pos] = i;
        pT += m[k];
    }

    if (t == 1023)
        nc_ws[b] = (T < MAXC) ? T : MAXC;
}

// ---------------------------------------------------------------------------
// Kernel 2: gather rows via CDNA5 async global<->LDS DMA (ASYNCcnt path).
//   Each block copies ROWS=4 rows of 4 KB. 256 threads * 16B = 4 KB per row.
//   Loads are issued back-to-back (4 in flight), one s_wait_asynccnt, then
//   stores are issued; s_endpgm's implicit wait-idle drains the stores.
// ---------------------------------------------------------------------------
__global__ __launch_bounds__(256)
void chunk_gather_kernel(const float* __restrict__ hid,       // [BB*LL*DD]
                         const int*   __restrict__ take_ws,   // [BB*MAXC]
                         const int*   __restrict__ nc_ws,     // [BB]
                         float* __restrict__ out_chunk,       // [BB*MAXC*DD]
                         float* __restrict__ out_pad,         // [BB*MAXC]
                         float* __restrict__ out_idx)         // [BB*MAXC]
{
    __shared__ __align__(16) float smem[ROWS * DD];           // 16 KB

    const int t    = threadIdx.x;                             // 0..255
    const int row0 = blockIdx.x * ROWS;                       // aligned, same batch
    const int b    = row0 >> 11;                              // row0 / MAXC
    const int nc   = nc_ws[b];

    const float* srcbase = hid + (size_t)b * (LL * DD);       // uniform saddr
    const unsigned lbase = (unsigned)(uintptr_t)(&smem[0]) + (unsigned)t * 16u;

    int  idxv[ROWS];
    bool valid[ROWS];
    #pragma unroll
    for (int r = 0; r < ROWS; ++r) {
        const int row = row0 + r;
        const int j   = row & (MAXC - 1);
        idxv[r]  = take_ws[row];
        valid[r] = (j < nc);
        if (t == 0) {
            out_pad[row] = valid[r] ? 1.0f : 0.0f;
            out_idx[row] = (float)idxv[r];
        }
    }

    // Issue gathered-row DMA loads: global -> LDS  (ASYNCcnt++)
    #pragma unroll
    for (int r = 0; r < ROWS; ++r) {
        if (valid[r]) {
            const unsigned voff  = (unsigned)idxv[r] * (DD * 4u) + (unsigned)t * 16u;
            const unsigned laddr = lbase + (unsigned)r * (DD * 4u);
            asm volatile("global_load_async_to_lds_b128 %0, %1, %2"
                         :: "v"(laddr), "v"(voff), "s"(srcbase) : "memory");
        }
    }

    // Wait for all async loads into LDS
    asm volatile("s_wait_asynccnt 0x0" ::: "memory");

    // Issue DMA stores: LDS -> global, or plain zero stores for padded rows
    #pragma unroll
    for (int r = 0; r < ROWS; ++r) {
        const unsigned row = (unsigned)(row0 + r);
        if (valid[r]) {
            const unsigned voff  = row * (DD * 4u) + (unsigned)t * 16u;
            const unsigned laddr = lbase + (unsigned)r * (DD * 4u);
            asm volatile("global_store_async_from_lds_b128 %0, %1, %2"
                         :: "v"(voff), "v"(laddr), "s"(out_chunk) : "memory");
        } else {
            float4 z = make_float4(0.f, 0.f, 0.f, 0.f);
            *(float4*)(out_chunk + (size_t)row * DD + (size_t)t * 4) = z;
        }
    }
    // s_endpgm performs implicit s_wait_idle -> async stores complete.
}

// ---------------------------------------------------------------------------
// Launcher
// ---------------------------------------------------------------------------
extern "C" void kernel_launch(void* const* d_in, const int* in_sizes, int n_in,
                              void* d_out, int out_size, void* d_ws, size_t ws_size,
                              hipStream_t stream)
{
    (void)in_sizes; (void)n_in; (void)out_size; (void)ws_size;

    const float* hid  = (const float*)d_in[0];   // hidden_states f32 [4,8192,1024]
    const int*   mask = (const int*)d_in[1];     // boundary_mask as int32 [4,8192]

    float* out       = (float*)d_out;
    float* out_chunk = out;                                   // [4,2048,1024]
    float* out_pad   = out + (size_t)BB * MAXC * DD;          // [4,2048]
    float* out_idx   = out_pad + (size_t)BB * MAXC;           // [4,2048]

    int* take_ws = (int*)d_ws;                                // [4*2048]
    int* nc_ws   = take_ws + (size_t)BB * MAXC;               // [4]

    chunk_scan_kernel<<<BB, 1024, 0, stream>>>(mask, take_ws, nc_ws);

    chunk_gather_kernel<<<(BB * MAXC) / ROWS, 256, 0, stream>>>(
        hid, take_ws, nc_ws, out_chunk, out_pad, out_idx);
}